// GCNModelVAE_52913997087388
// MI455X (gfx1250) — compile-verified
//
#include <hip/hip_runtime.h>

typedef __attribute__((ext_vector_type(2))) float v2f;
typedef __attribute__((ext_vector_type(8))) float v8f;

#define N_NODES 10000
#define N_EDGES 320000
#define D_IN    1024
#define H1      64
#define H2      32

// ---------------------------------------------------------------------------
// 0. zero-fill (atomic accumulators must be re-zeroed every launch/replay)
// ---------------------------------------------------------------------------
__global__ void zero_f32(float* __restrict__ p, int n) {
  int i = blockIdx.x * blockDim.x + threadIdx.x;
  if (i < n) p[i] = 0.0f;
}

// ---------------------------------------------------------------------------
// 1. h0 = x @ W0   [10000,1024] x [1024,64] -> [10000,64]
//    one wave per 16x16 output tile, V_WMMA_F32_16X16X4_F32, K-loop of 256
// ---------------------------------------------------------------------------
__global__ void gemm_x_w0(const float* __restrict__ x,
                          const float* __restrict__ W0,
                          float* __restrict__ h0) {
  const int tiles_m = N_NODES / 16;   // 625
  const int tiles_n = H1 / 16;        // 4
  int wave = (blockIdx.x * blockDim.x + threadIdx.x) >> 5;
  int lane = threadIdx.x & 31;
  if (wave >= tiles_m * tiles_n) return;          // wave-uniform: EXEC stays full
  int ti = wave / tiles_n;
  int tj = wave % tiles_n;

  int m  = lane & 15;               // A row / B col within tile
  int kr = (lane >> 4) << 1;        // K sub-offset {0,2} per half-wave

  const float* arow = x  + (size_t)(ti * 16 + m) * D_IN;
  const float* bcol = W0 + (tj * 16 + m);          // B[k][n] = W0[k*H1 + n]

  v8f c = {};
  #pragma unroll 4
  for (int kb = 0; kb < D_IN; kb += 4) {
    v2f a = *(const v2f*)(arow + kb + kr);         // A[m][kb+kr], A[m][kb+kr+1]
    v2f b;
    b.x = bcol[(size_t)(kb + kr) * H1];
    b.y = bcol[(size_t)(kb + kr + 1) * H1];
    c = __builtin_amdgcn_wmma_f32_16x16x4_f32(false, a, false, b,
                                              (short)0, c, false, false);
  }

  int row0 = ti * 16 + (lane >> 4) * 8;
  int col  = tj * 16 + (lane & 15);
  #pragma unroll
  for (int r = 0; r < 8; ++r)
    h0[(size_t)(row0 + r) * H1 + col] = c[r];
}

// ---------------------------------------------------------------------------
// 2. agg1[dst] += w * h0[src]   (thread per (edge, dim) -> coalesced reads)
// ---------------------------------------------------------------------------
__global__ void edge_agg_h(const float* __restrict__ h0,
                           const int*  __restrict__ src,
                           const int*  __restrict__ dst,
                           const float* __restrict__ w,
                           float* __restrict__ agg1) {
  long long tid = (long long)blockIdx.x * blockDim.x + threadIdx.x;
  if (tid >= (long long)N_EDGES * H1) return;
  int e = (int)(tid >> 6);
  int d = (int)(tid & 63);
  float v = w[e] * h0[(size_t)src[e] * H1 + d];
  unsafeAtomicAdd(&agg1[(size_t)dst[e] * H1 + d], v);
}

// ---------------------------------------------------------------------------
// 3. m1 = relu(agg1) @ W1, m2 = relu(agg1) @ W2   [10000,64]x[64,32]
//    one wave per 16x16 tile of either output; ReLU fused into A load
// ---------------------------------------------------------------------------
__global__ void gemm_h1_w(const float* __restrict__ agg1,
                          const float* __restrict__ W1,
                          const float* __restrict__ W2,
                          float* __restrict__ m1,
                          float* __restrict__ m2) {
  const int tiles_m = N_NODES / 16;   // 625
  const int tiles_n = H2 / 16;        // 2
  int wave = (blockIdx.x * blockDim.x + threadIdx.x) >> 5;
  int lane = threadIdx.x & 31;
  if (wave >= tiles_m * tiles_n * 2) return;
  int which = wave & 1;
  int t  = wave >> 1;
  int ti = t / tiles_n;
  int tj = t % tiles_n;
  const float* W   = which ? W2 : W1;
  float*       out = which ? m2 : m1;

  int m  = lane & 15;
  int kr = (lane >> 4) << 1;
  const float* arow = agg1 + (size_t)(ti * 16 + m) * H1;
  const float* bcol = W + (tj * 16 + m);

  v8f c = {};
  #pragma unroll
  for (int kb = 0; kb < H1; kb += 4) {
    v2f a;
    a.x = fmaxf(arow[kb + kr],     0.0f);   // h1 = relu(agg1)
    a.y = fmaxf(arow[kb + kr + 1], 0.0f);
    v2f b;
    b.x = bcol[(size_t)(kb + kr) * H2];
    b.y = bcol[(size_t)(kb + kr + 1) * H2];
    c = __builtin_amdgcn_wmma_f32_16x16x4_f32(false, a, false, b,
                                              (short)0, c, false, false);
  }

  int row0 = ti * 16 + (lane >> 4) * 8;
  int col  = tj * 16 + (lane & 15);
  #pragma unroll
  for (int r = 0; r < 8; ++r)
    out[(size_t)(row0 + r) * H2 + col] = c[r];
}

// ---------------------------------------------------------------------------
// 4. aggm1[dst] += w*m1[src] ; aggm2[dst] += w*m2[src]   (both in one pass)
// ---------------------------------------------------------------------------
__global__ void edge_agg_z(const float* __restrict__ m1,
                           const float* __restrict__ m2,
                           const int*  __restrict__ src,
                           const int*  __restrict__ dst,
                           const float* __restrict__ w,
                           float* __restrict__ aggm1,
                           float* __restrict__ aggm2) {
  long long tid = (long long)blockIdx.x * blockDim.x + threadIdx.x;
  if (tid >= (long long)N_EDGES * 64) return;
  int e = (int)(tid >> 6);
  int q = (int)(tid & 63);
  float wt = w[e];
  int s = src[e], dd = dst[e];
  if (q < H2) {
    unsafeAtomicAdd(&aggm1[(size_t)dd * H2 + q], wt * m1[(size_t)s * H2 + q]);
  } else {
    int q2 = q - H2;
    unsafeAtomicAdd(&aggm2[(size_t)dd * H2 + q2], wt * m2[(size_t)s * H2 + q2]);
  }
}

// ---------------------------------------------------------------------------
// 5. z = z_mean + eps * exp(z_log_std)
// ---------------------------------------------------------------------------
__global__ void reparam(const float* __restrict__ aggm1,
                        const float* __restrict__ aggm2,
                        const float* __restrict__ eps,
                        float* __restrict__ z) {
  int i = blockIdx.x * blockDim.x + threadIdx.x;
  if (i < N_NODES * H2) {
    z[i] = aggm1[i] + eps[i] * __expf(aggm2[i]);
  }
}

// ---------------------------------------------------------------------------
// 6. recon = z @ z.T   [10000,32] -> [10000,10000]  (store-bound: 400 MB out)
//    one wave per 16x16 tile; 8x V_WMMA_F32_16X16X4_F32 (K = 32)
//    z is 1.28 MB -> resident in L2 (192 MB) for all 390625 tiles
// ---------------------------------------------------------------------------
__global__ void decoder_zzT(const float* __restrict__ z,
                            float* __restrict__ out) {
  const int T = N_NODES / 16;  // 625
  int wave = (blockIdx.x * blockDim.x + threadIdx.x) >> 5;
  int lane = threadIdx.x & 31;
  if (wave >= T * T) return;
  int ti = wave / T;
  int tj = wave % T;

  int m  = lane & 15;
  int kr = (lane >> 4) << 1;
  const float* arow = z + (size_t)(ti * 16 + m) * H2;  // A[m][k]  = z[ti*16+m][k]
  const float* brow = z + (size_t)(tj * 16 + m) * H2;  // B[k][n]  = z[tj*16+n][k], n==m per lane

  v8f c = {};
  #pragma unroll
  for (int kb = 0; kb < H2; kb += 4) {
    v2f a = *(const v2f*)(arow + kb + kr);
    v2f b = *(const v2f*)(brow + kb + kr);
    c = __builtin_amdgcn_wmma_f32_16x16x4_f32(false, a, false, b,
                                              (short)0, c, false, false);
  }

  int row0 = ti * 16 + (lane >> 4) * 8;
  int col  = tj * 16 + (lane & 15);
  #pragma unroll
  for (int r = 0; r < 8; ++r)
    out[(size_t)(row0 + r) * N_NODES + col] = c[r];
}

// ---------------------------------------------------------------------------
// launch
// ---------------------------------------------------------------------------
extern "C" void kernel_launch(void* const* d_in, const int* in_sizes, int n_in,
                              void* d_out, int out_size, void* d_ws, size_t ws_size,
                              hipStream_t stream) {
  const float* x    = (const float*)d_in[0];
  const int*   esrc = (const int*)  d_in[1];
  const int*   edst = (const int*)  d_in[2];
  const float* ew   = (const float*)d_in[3];
  const float* W0   = (const float*)d_in[4];
  const float* W1   = (const float*)d_in[5];
  const float* W2   = (const float*)d_in[6];
  const float* eps  = (const float*)d_in[7];
  float* out = (float*)d_out;

  float* ws    = (float*)d_ws;
  float* h0    = ws;                       // 640000 floats
  float* agg1  = ws + 640000;              // 640000
  float* m1    = ws + 1280000;             // 320000
  float* m2    = ws + 1600000;             // 320000
  float* aggm1 = ws + 1920000;             // 320000
  float* aggm2 = ws + 2240000;             // 320000
  float* z     = ws + 2560000;             // 320000  (total 2.88M floats = 11.52 MB)

  // zero all accumulators / scratch (harness poisons ws; atomics need zeros)
  {
    int n = 2880000;
    zero_f32<<<(n + 255) / 256, 256, 0, stream>>>(ws, n);
  }
  // h0 = x @ W0
  {
    int waves = (N_NODES / 16) * (H1 / 16);          // 2500
    gemm_x_w0<<<(waves * 32 + 255) / 256, 256, 0, stream>>>(x, W0, h0);
  }
  // agg1 = scatter-add
  {
    long long thr = (long long)N_EDGES * H1;         // 20.48M
    edge_agg_h<<<(int)((thr + 255) / 256), 256, 0, stream>>>(h0, esrc, edst, ew, agg1);
  }
  // m1 = relu(agg1)@W1, m2 = relu(agg1)@W2
  {
    int waves = (N_NODES / 16) * (H2 / 16) * 2;      // 2500
    gemm_h1_w<<<(waves * 32 + 255) / 256, 256, 0, stream>>>(agg1, W1, W2, m1, m2);
  }
  // aggm1 / aggm2 = scatter-add
  {
    long long thr = (long long)N_EDGES * 64;         // 20.48M
    edge_agg_z<<<(int)((thr + 255) / 256), 256, 0, stream>>>(m1, m2, esrc, edst, ew,
                                                             aggm1, aggm2);
  }
  // z = z_mean + eps * exp(z_log_std)
  {
    int n = N_NODES * H2;
    reparam<<<(n + 255) / 256, 256, 0, stream>>>(aggm1, aggm2, eps, z);
  }
  // recon = z @ z.T
  {
    long long waves = (long long)(N_NODES / 16) * (N_NODES / 16);  // 390625
    decoder_zzT<<<(int)((waves * 32 + 255) / 256), 256, 0, stream>>>(z, out);
  }
}